// st_attention_block_40819369181195
// MI455X (gfx1250) — compile-verified
//
#include <hip/hip_runtime.h>
#include <hip/hip_bf16.h>
#include <math.h>

typedef __attribute__((ext_vector_type(2))) float v2f;
typedef __attribute__((ext_vector_type(8))) float v8f;

#define Nn 256
#define Cc 64
#define Tt 128
#define Vv 25
#define Oo 64
#define Rr 4
#define Ss (Tt * Vv)   // 3200, contiguous (t,v) plane per (n,c)

// ---------------------------------------------------------------------------
// Kernel 1: v[n] = Wv(64x64) @ x[n](64x3200) + b  via V_WMMA_F32_16X16X4_F32.
// WG = 256 threads = 8 waves. Waves tile M(=O, 4 tiles of 16) x S(2 tiles of 16).
// x block [64 x 32] staged in LDS (coalesced), K loop of 16 WMMA steps (K=4 each).
// ---------------------------------------------------------------------------
__global__ __launch_bounds__(256)
void vproj_kernel(const float* __restrict__ x, const float* __restrict__ Wv_w,
                  const float* __restrict__ Wv_b, float* __restrict__ vout)
{
    __shared__ float xs[Cc * 32];
    const int n   = blockIdx.y;
    const int sb  = blockIdx.x * 32;
    const int tid = threadIdx.x;
    const float* xn = x + (size_t)n * Cc * Ss;

    #pragma unroll
    for (int i = 0; i < 8; ++i) {
        int idx = tid + i * 256;
        int c = idx >> 5, sc = idx & 31;
        xs[idx] = xn[(size_t)c * Ss + sb + sc];
    }
    __syncthreads();

    const int lane  = tid & 31;
    const int wave  = tid >> 5;
    const int mtile = (wave & 3) * 16;          // O tile
    const int sg    = (wave >> 2) * 16;         // S sub-tile within WG block
    const int m     = mtile + (lane & 15);
    const int kh    = (lane >> 4) << 1;         // 0 or 2 (K sub-select per ISA layout)
    const int scol  = sg + (lane & 15);

    v8f acc = {};
    const float* wrow = Wv_w + m * Cc;          // A: Wv row-major [O][C]
    #pragma unroll 4
    for (int kb = 0; kb < Cc; kb += 4) {
        v2f a, b;
        a.x = wrow[kb + kh];
        a.y = wrow[kb + kh + 1];
        b.x = xs[(kb + kh) * 32 + scol];
        b.y = xs[(kb + kh + 1) * 32 + scol];
        acc = __builtin_amdgcn_wmma_f32_16x16x4_f32(false, a, false, b,
                                                    (short)0, acc, false, false);
    }

    const int sgl = sb + scol;
    float* vn = vout + (size_t)n * Oo * Ss;
    #pragma unroll
    for (int i = 0; i < 8; ++i) {
        int o = mtile + i + ((lane >> 4) << 3);   // D layout: M = i + 8*(lane/16)
        vn[(size_t)o * Ss + sgl] = acc[i] + Wv_b[o];
    }
}

// ---------------------------------------------------------------------------
// Kernel 2: xbar[c][v] = sum_t x[n,c,t,v] in LDS, then k/q = W·xbar/T + b.
// One WG per n. Tiny output (k,q are 100 floats each per n).
// ---------------------------------------------------------------------------
__global__ __launch_bounds__(256)
void kq_kernel(const float* __restrict__ x,
               const float* __restrict__ Wk_w, const float* __restrict__ Wk_b,
               const float* __restrict__ Wq_w, const float* __restrict__ Wq_b,
               float* __restrict__ kout, float* __restrict__ qout)
{
    __shared__ float xb[Cc * Vv];   // 1600 floats
    const int n   = blockIdx.x;
    const int tid = threadIdx.x;
    const float* xn = x + (size_t)n * Cc * Ss;

    for (int p = tid; p < Cc * Vv; p += 256) {
        int c = p / Vv, v = p % Vv;
        const float* base = xn + (size_t)c * Ss + v;
        float s = 0.f;
        #pragma unroll 4
        for (int t = 0; t < Tt; ++t) s += base[t * Vv];
        xb[p] = s;
    }
    __syncthreads();

    if (tid < 200) {
        int which = tid / 100;
        int p = tid % 100;
        int r = p / Vv, v = p % Vv;
        const float* W = which ? Wq_w : Wk_w;
        const float* B = which ? Wq_b : Wk_b;
        float s = 0.f;
        #pragma unroll
        for (int c = 0; c < Cc; ++c) s += W[r * Cc + c] * xb[c * Vv + v];
        s = s * (1.0f / (float)Tt) + B[r];
        (which ? qout : kout)[n * 100 + p] = s;
    }
}

// ---------------------------------------------------------------------------
// Kernel 3: att[n,o,v,u] = alpha*(ffn(tanh([conv1(k⊗q); conv2(k⊖q)]))) + adjc.
// Tiny MLP (8->64) per (v,u): VALU, one WG per n, all constants in LDS.
// ---------------------------------------------------------------------------
__global__ __launch_bounds__(256)
void att_kernel(const float* __restrict__ kbuf, const float* __restrict__ qbuf,
                const float* __restrict__ adjc,
                const float* __restrict__ c1w, const float* __restrict__ c1b,
                const float* __restrict__ c2w, const float* __restrict__ c2b,
                const float* __restrict__ fw,  const float* __restrict__ fb,
                const float* __restrict__ alpha, float* __restrict__ att)
{
    __shared__ float kL[100], qL[100], adjL[Vv * Vv];
    __shared__ float fwL[Oo * 8], fbL[Oo];
    __shared__ float w1[16], w2[16], b1[4], b2[4];
    const int n   = blockIdx.x;
    const int tid = threadIdx.x;

    if (tid < 100) { kL[tid] = kbuf[n * 100 + tid]; qL[tid] = qbuf[n * 100 + tid]; }
    for (int p = tid; p < Vv * Vv; p += 256) adjL[p] = adjc[p];
    for (int p = tid; p < Oo * 8;  p += 256) fwL[p]  = fw[p];
    if (tid < Oo) fbL[tid] = fb[tid];
    if (tid < 16) { w1[tid] = c1w[tid]; w2[tid] = c2w[tid]; }
    if (tid < 4)  { b1[tid] = c1b[tid]; b2[tid] = c2b[tid]; }
    __syncthreads();

    const float al = alpha[0];
    float* attn = att + (size_t)n * Oo * Vv * Vv;
    for (int p = tid; p < Vv * Vv; p += 256) {
        int v = p / Vv, u = p % Vv;
        float x1[4], x2[4], a[8];
        #pragma unroll
        for (int r = 0; r < 4; ++r) {
            float kv = kL[r * Vv + v], qu = qL[r * Vv + u];
            x1[r] = kv * qu;
            x2[r] = kv - qu;
        }
        #pragma unroll
        for (int s = 0; s < 4; ++s) {
            float y1 = b1[s], y2 = b2[s];
            #pragma unroll
            for (int r = 0; r < 4; ++r) { y1 += w1[s * 4 + r] * x1[r]; y2 += w2[s * 4 + r] * x2[r]; }
            a[s]     = tanhf(y1);
            a[4 + s] = tanhf(y2);
        }
        float adj = adjL[p];
        for (int o = 0; o < Oo; ++o) {
            float acc = fbL[o];
            #pragma unroll
            for (int j = 0; j < 8; ++j) acc += fwL[o * 8 + j] * a[j];
            attn[(size_t)o * (Vv * Vv) + p] = al * acc + adj;
        }
    }
}

// ---------------------------------------------------------------------------
// Kernel 4: out[n,o] (128x25) = v[n,o] (128xU) @ att[n,o]^T (Ux25), WMMA f32.
// One WG per (n,o). v tile [128x28] and att [32x28] zero-padded in LDS.
// Wave w owns T rows [16w,16w+16); two N tiles (v cols 0..15, 16..24).
// 7 K-steps of 4 over U (25..27 padded with zeros) -> 14 WMMA per wave.
// ---------------------------------------------------------------------------
__global__ __launch_bounds__(256)
void out_kernel(const float* __restrict__ vbuf, const float* __restrict__ att,
                float* __restrict__ out)
{
    __shared__ float attS[32 * 28];
    __shared__ float vS[128 * 28];
    const int o   = blockIdx.x;
    const int n   = blockIdx.y;
    const int tid = threadIdx.x;
    const size_t no = (size_t)n * Oo + o;
    const float* vno = vbuf + no * (Tt * Vv);
    const float* ano = att  + no * (Vv * Vv);

    for (int p = tid; p < 32 * 28;  p += 256) attS[p] = 0.f;
    for (int p = tid; p < 128 * 28; p += 256) vS[p]   = 0.f;
    __syncthreads();
    for (int p = tid; p < Vv * Vv; p += 256) {
        int v = p / Vv, u = p % Vv;
        attS[v * 28 + u] = ano[p];          // B^T source: B[u][v] = att[v][u]
    }
    for (int p = tid; p < Tt * Vv; p += 256) {
        int t = p / Vv, u = p % Vv;
        vS[t * 28 + u] = vno[p];
    }
    __syncthreads();

    const int lane  = tid & 31;
    const int wave  = tid >> 5;
    const int tbase = wave * 16;
    const int kh    = (lane >> 4) << 1;
    const int ml    = lane & 15;

    v8f acc0 = {}, acc1 = {};
    #pragma unroll
    for (int kb = 0; kb < 28; kb += 4) {
        int u0 = kb + kh;
        v2f a, b0, b1;
        a.x  = vS[(tbase + ml) * 28 + u0];
        a.y  = vS[(tbase + ml) * 28 + u0 + 1];
        b0.x = attS[ml * 28 + u0];            // N tile 0: v cols 0..15
        b0.y = attS[ml * 28 + u0 + 1];
        b1.x = attS[(16 + ml) * 28 + u0];     // N tile 1: v cols 16..31 (25+ are zero)
        b1.y = attS[(16 + ml) * 28 + u0 + 1];
        acc0 = __builtin_amdgcn_wmma_f32_16x16x4_f32(false, a, false, b0,
                                                     (short)0, acc0, false, false);
        acc1 = __builtin_amdgcn_wmma_f32_16x16x4_f32(false, a, false, b1,
                                                     (short)0, acc1, false, false);
    }

    float* ot = out + no * (Tt * Vv);
    #pragma unroll
    for (int i = 0; i < 8; ++i) {
        int t  = tbase + i + ((lane >> 4) << 3);
        int v1 = 16 + ml;
        ot[t * Vv + ml] = acc0[i];
        if (v1 < Vv) ot[t * Vv + v1] = acc1[i];
    }
}

// ---------------------------------------------------------------------------
extern "C" void kernel_launch(void* const* d_in, const int* in_sizes, int n_in,
                              void* d_out, int out_size, void* d_ws, size_t ws_size,
                              hipStream_t stream)
{
    const float* x     = (const float*)d_in[0];
    const float* adjc  = (const float*)d_in[1];
    const float* Wk_w  = (const float*)d_in[2];
    const float* Wk_b  = (const float*)d_in[3];
    const float* Wq_w  = (const float*)d_in[4];
    const float* Wq_b  = (const float*)d_in[5];
    const float* Wv_w  = (const float*)d_in[6];
    const float* Wv_b  = (const float*)d_in[7];
    const float* c1w   = (const float*)d_in[8];
    const float* c1b   = (const float*)d_in[9];
    const float* c2w   = (const float*)d_in[10];
    const float* c2b   = (const float*)d_in[11];
    const float* fw    = (const float*)d_in[12];
    const float* fb    = (const float*)d_in[13];
    const float* alpha = (const float*)d_in[14];
    float* out = (float*)d_out;

    float* ws   = (float*)d_ws;
    float* vbuf = ws;                                       // N*O*T*V floats
    float* attb = vbuf + (size_t)Nn * Oo * Tt * Vv;         // N*O*V*V floats
    float* kbuf = attb + (size_t)Nn * Oo * Vv * Vv;         // N*R*V floats
    float* qbuf = kbuf + (size_t)Nn * Rr * Vv;

    vproj_kernel<<<dim3(Ss / 32, Nn), 256, 0, stream>>>(x, Wv_w, Wv_b, vbuf);
    kq_kernel<<<dim3(Nn), 256, 0, stream>>>(x, Wk_w, Wk_b, Wq_w, Wq_b, kbuf, qbuf);
    att_kernel<<<dim3(Nn), 256, 0, stream>>>(kbuf, qbuf, adjc, c1w, c1b,
                                             c2w, c2b, fw, fb, alpha, attb);
    out_kernel<<<dim3(Oo, Nn), 256, 0, stream>>>(vbuf, attb, out);
}